// SRR_40080634806832
// MI455X (gfx1250) — compile-verified
//
#include <hip/hip_runtime.h>
#include <math.h>

typedef float v2f __attribute__((ext_vector_type(2)));
typedef float v8f __attribute__((ext_vector_type(8)));

#define HC 128  // hidden channels

// ---------------- utility fills ----------------
__global__ void k_fill_f32(float* __restrict__ p, long n) {
  long j = ((long)blockIdx.x * blockDim.x + threadIdx.x) * 4;
  if (j + 3 < n) {
    *(float4*)(p + j) = make_float4(0.f, 0.f, 0.f, 0.f);
  } else {
    for (int t = 0; t < 4; ++t) if (j + t < n) p[j + t] = 0.f;
  }
}

__global__ void k_fill_i32(int* __restrict__ p, int n) {
  int i = blockIdx.x * blockDim.x + threadIdx.x;
  if (i < n) p[i] = 0;
}

// ---------------- degree / norm precompute ----------------
__global__ void k_degcount(const int* __restrict__ dst, int E, int* __restrict__ degi) {
  int e = blockIdx.x * blockDim.x + threadIdx.x;
  if (e < E) atomicAdd(&degi[dst[e]], 1);
}

__global__ void k_dinv(const int* __restrict__ degi, int NN, float alpha,
                       float* __restrict__ dinv, float* __restrict__ selfc) {
  int n = blockIdx.x * blockDim.x + threadIdx.x;
  if (n < NN) {
    float d = (float)(degi[n] + 1);  // +1 self loop
    float r = rsqrtf(d);
    dinv[n] = r;
    selfc[n] = (1.0f - alpha) * r * r;  // self-loop coefficient, pre-scaled by (1-alpha)
  }
}

__global__ void k_coef(const int* __restrict__ src, const int* __restrict__ dst, int E,
                       const float* __restrict__ dinv, float alpha, float* __restrict__ coef) {
  int e = blockIdx.x * blockDim.x + threadIdx.x;
  if (e < E) coef[e] = (1.0f - alpha) * dinv[src[e]] * dinv[dst[e]];
}

// ---------------- SpMM: agg[dst] += coef * h[src]  (one wave32 per edge) ----------------
__global__ __launch_bounds__(256) void k_spmm(const int* __restrict__ src,
                                              const int* __restrict__ dst,
                                              const float* __restrict__ coef,
                                              const float* __restrict__ h,
                                              float* __restrict__ agg, int E) {
  long tid = (long)blockIdx.x * blockDim.x + threadIdx.x;
  int e = (int)(tid >> 5);
  int lane = (int)(tid & 31);
  if (e >= E) return;

  // prefetch a future edge's source row into cache (global_prefetch_b8)
  int e2 = e + 1024;
  if (e2 < E) {
    int s2 = src[e2];
    __builtin_prefetch(h + (long)s2 * HC + lane * 4, 0, 3);
  }

  int s = src[e];
  int d = dst[e];
  float c = coef[e];
  const float4 hv = *(const float4*)(h + (long)s * HC + lane * 4);
  float* ap = agg + (long)d * HC + lane * 4;
  unsafeAtomicAdd(ap + 0, c * hv.x);
  unsafeAtomicAdd(ap + 1, c * hv.y);
  unsafeAtomicAdd(ap + 2, c * hv.z);
  unsafeAtomicAdd(ap + 3, c * hv.w);
}

// ---------------- WMMA fp32 GEMM (fused GCNII update) ----------------
// MODE 0 (proj):  out = A @ W + bias
// MODE 1 (layer): z   = A(=agg) + selfc*hprev + alpha*x0   (built in the A-fragment load)
//                 out = z @ (beta*W + (1-beta)*I) + hprev   [ReLU if do_relu]
//
// One block = 256 threads = 8 waves; block covers 128 rows; each wave computes a
// 16x128 strip as 8 accumulator tiles of 16x16 via V_WMMA_F32_16X16X4_F32.
// W' staged in LDS as [k/4][n][k%4] so a B-frag is one conflict-free b64 load.
template <int MODE>
__global__ __launch_bounds__(256) void k_gemm(const float* __restrict__ A,
                                              const float* __restrict__ W,
                                              const float* __restrict__ bias,
                                              const float* __restrict__ hprev,
                                              const float* __restrict__ x0,
                                              const float* __restrict__ selfc,
                                              float* __restrict__ out,
                                              int n_rows, float beta, float alpha,
                                              int do_relu) {
  __shared__ float lds[32 * HC * 4];  // 64 KB: [k>>2][n][k&3]

  const int t = threadIdx.x;
  // stage (possibly modified) weight matrix into LDS
  for (int it = 0; it < (HC * HC) / 256; ++it) {
    int idx = it * 256 + t;
    int k = idx >> 7;
    int n = idx & (HC - 1);
    float w = W[idx];
    if (MODE == 1) w = beta * w + ((k == n) ? (1.0f - beta) : 0.0f);
    lds[(((k >> 2) * HC) + n) * 4 + (k & 3)] = w;
  }
  __syncthreads();

  const int wave = t >> 5;
  const int lane = t & 31;
  const int half = lane >> 4;  // 0: lanes 0-15, 1: lanes 16-31
  const int r = lane & 15;
  const int m0 = blockIdx.x * 128 + wave * 16;

  int rowL = m0 + r;
  if (rowL >= n_rows) rowL = n_rows - 1;  // clamp: EXEC must stay all-1 for WMMA
  const long rbase = (long)rowL * HC;
  const float sc = (MODE == 1) ? selfc[rowL] : 0.0f;

  v8f acc[8];
#pragma unroll
  for (int nt = 0; nt < 8; ++nt) acc[nt] = (v8f){0.f, 0.f, 0.f, 0.f, 0.f, 0.f, 0.f, 0.f};

  for (int kk = 0; kk < HC; kk += 4) {
    const int kc = kk + 2 * half;  // A frag: VGPR0 holds K=kc, VGPR1 holds K=kc+1
    v2f a;
    if (MODE == 0) {
      const float2 av = *(const float2*)(A + rbase + kc);
      a.x = av.x;
      a.y = av.y;
    } else {
      const float2 g = *(const float2*)(A + rbase + kc);
      const float2 hh = *(const float2*)(hprev + rbase + kc);
      const float2 xx = *(const float2*)(x0 + rbase + kc);
      a.x = g.x + sc * hh.x + alpha * xx.x;
      a.y = g.y + sc * hh.y + alpha * xx.y;
    }
    const float* bp = lds + ((kk >> 2) * HC) * 4 + 2 * half;
#pragma unroll
    for (int nt = 0; nt < 8; ++nt) {
      const float2 bv = *(const float2*)(bp + (nt * 16 + r) * 4);
      v2f b;
      b.x = bv.x;
      b.y = bv.y;
      acc[nt] = __builtin_amdgcn_wmma_f32_16x16x4_f32(
          /*neg_a=*/false, a, /*neg_b=*/false, b,
          /*c_mod=*/(short)0, acc[nt], /*reuse_a=*/false, /*reuse_b=*/false);
    }
  }

  // D tile: VGPR v -> row m0 + v + 8*half, col nt*16 + r
#pragma unroll
  for (int nt = 0; nt < 8; ++nt) {
    const int col = nt * 16 + r;
#pragma unroll
    for (int v = 0; v < 8; ++v) {
      const int row = m0 + v + 8 * half;
      if (row < n_rows) {
        float val = acc[nt][v];
        if (MODE == 0) {
          val += bias[col];
        } else {
          val += hprev[(long)row * HC + col];
          if (do_relu) val = fmaxf(val, 0.0f);
        }
        out[(long)row * HC + col] = val;
      }
    }
  }
}

// ---------------- host orchestration ----------------
extern "C" void kernel_launch(void* const* d_in, const int* in_sizes, int n_in,
                              void* d_out, int out_size, void* d_ws, size_t ws_size,
                              hipStream_t stream) {
  const float* x = (const float*)d_in[0];
  const int* eidx = (const int*)d_in[1];  // [2, E]
  const float* W_proj = (const float*)d_in[2];
  const float* b_proj = (const float*)d_in[3];
  const float* conv_w = (const float*)d_in[4];  // [8, 128, 128]

  const int NN = in_sizes[0] / HC;
  const int E = in_sizes[1] / 2;
  const int* src = eidx;
  const int* dst = eidx + E;
  const float ALPHA = 0.1f;
  const float THETA = 0.5f;

  // workspace carve
  char* ws = (char*)d_ws;
  float* x0 = (float*)ws;    ws += (size_t)NN * HC * sizeof(float);
  float* hA = (float*)ws;    ws += (size_t)NN * HC * sizeof(float);
  float* agg = (float*)ws;   ws += (size_t)NN * HC * sizeof(float);
  float* coef = (float*)ws;  ws += (size_t)E * sizeof(float);
  float* dinv = (float*)ws;  ws += (size_t)NN * sizeof(float);
  float* selfc = (float*)ws; ws += (size_t)NN * sizeof(float);
  int* degi = (int*)ws;      ws += (size_t)NN * sizeof(int);
  float* O = (float*)d_out;

  // degrees + symmetric norm (with self-loops)
  k_fill_i32<<<(NN + 255) / 256, 256, 0, stream>>>(degi, NN);
  k_degcount<<<(E + 255) / 256, 256, 0, stream>>>(dst, E, degi);
  k_dinv<<<(NN + 255) / 256, 256, 0, stream>>>(degi, NN, ALPHA, dinv, selfc);
  k_coef<<<(E + 255) / 256, 256, 0, stream>>>(src, dst, E, dinv, ALPHA, coef);

  const int gblk = (NN + 127) / 128;
  // x0 = x @ W_proj + b_proj
  k_gemm<0><<<gblk, 256, 0, stream>>>(x, W_proj, b_proj, nullptr, nullptr, nullptr,
                                      x0, NN, 0.f, 0.f, 0);

  const long agg_n = (long)NN * HC;
  const int fill_blocks = (int)((agg_n / 4 + 255) / 256);
  const long spmm_threads = (long)E * 32;
  const int spmm_blocks = (int)((spmm_threads + 255) / 256);

  for (int i = 0; i < 8; ++i) {
    const float beta = logf(THETA / (float)(i + 1) + 1.0f);
    const float* hin = (i == 0) ? x0 : ((i & 1) ? hA : O);
    float* hout = (i & 1) ? O : hA;  // layer 7 (last, odd) lands in d_out

    k_fill_f32<<<fill_blocks, 256, 0, stream>>>(agg, agg_n);
    k_spmm<<<spmm_blocks, 256, 0, stream>>>(src, dst, coef, hin, agg, E);
    k_gemm<1><<<gblk, 256, 0, stream>>>(agg, conv_w + (size_t)i * HC * HC, nullptr,
                                        hin, x0, selfc, hout, NN, beta, ALPHA,
                                        (i != 7) ? 1 : 0);
  }
}